// LSTMSmallNetwork_4183298146953
// MI455X (gfx1250) — compile-verified
//
#include <hip/hip_runtime.h>

#define BATCH 1024
#define SEQT  512
#define HDIM  10
#define GDIM  40   // 4*H

typedef __attribute__((ext_vector_type(2))) float v2f;
typedef __attribute__((ext_vector_type(8))) float v8f;

__device__ __forceinline__ float fast_sigmoid(float v) {
    return __builtin_amdgcn_rcpf(1.f + __expf(-v));
}
__device__ __forceinline__ float fast_tanh(float v) {
    float e = __expf(-2.f * v);
    return (1.f - e) * __builtin_amdgcn_rcpf(1.f + e);
}

// One wave (32 threads) per (16-batch tile, direction). fp32 WMMA 16x16x4.
// A operand (16x4 f32): lane m (0-15) holds K=kb,kb+1 ; lane m+16 holds K=kb+2,kb+3.
// B operand (4x16 f32): lane n (0-15) holds K=kb,kb+1 rows at col n; lanes 16-31 K=kb+2,kb+3.
// C/D (16x16 f32): vgpr j, lanes 0-15 -> (M=j,     N=lane),
//                          lanes 16-31 -> (M=j+8, N=lane-16).
__global__ __launch_bounds__(32) void lstm_l0_scan(
    const float* __restrict__ x,
    const float* __restrict__ wih_f, const float* __restrict__ whh_f,
    const float* __restrict__ bih_f, const float* __restrict__ bhh_f,
    const float* __restrict__ wih_r, const float* __restrict__ whh_r,
    const float* __restrict__ bih_r, const float* __restrict__ bhh_r,
    float* __restrict__ l0f, float* __restrict__ l0b)
{
    const int dir = blockIdx.y;
    const float* wih = dir ? wih_r : wih_f;   // [40,1]
    const float* whh = dir ? whh_r : whh_f;   // [40,10]
    const float* bih = dir ? bih_r : bih_f;
    const float* bhh = dir ? bhh_r : bhh_f;
    float* outp = dir ? l0b : l0f;            // [B,T,10]

    const int b0   = blockIdx.x * 16;
    const int lane = threadIdx.x;
    const int mrow = lane & 15;
    const int half = lane >> 4;

    __shared__ __align__(16) float h_lds[16 * 12];   // h state, K-padded to 12
    __shared__ __align__(32) float g_lds[32 * 24];   // gate staging (acc layout)

    for (int i = lane; i < 16 * 12; i += 32) h_lds[i] = 0.f;

    // Preload B tiles of whh^T, plus per-column input weight & fused bias.
    v2f bwh[3][3];
    float wcol[3], bcol[3];
    #pragma unroll
    for (int tile = 0; tile < 3; ++tile) {
        int col = tile * 16 + mrow;
        int cv  = (col < GDIM);
        wcol[tile] = cv ? wih[col] : 0.f;
        bcol[tile] = cv ? (bih[col] + bhh[col]) : 0.f;
        #pragma unroll
        for (int ks = 0; ks < 3; ++ks) {
            int kb = ks * 4 + 2 * half;
            v2f bb; bb.x = 0.f; bb.y = 0.f;
            if (cv && kb < HDIM)     bb.x = whh[col * HDIM + kb];
            if (cv && kb + 1 < HDIM) bb.y = whh[col * HDIM + kb + 1];
            bwh[tile][ks] = bb;
        }
    }

    // Each lane owns 5 (row m, hidden hidx) pairs for the gate nonlinearity.
    int pm[5], ph[5];
    float cst[5], hnew[5];
    #pragma unroll
    for (int p = 0; p < 5; ++p) {
        int q = lane + 32 * p;
        pm[p] = q / HDIM; ph[p] = q % HDIM;
        cst[p] = 0.f;
    }

    __syncthreads();

    for (int step = 0; step < SEQT; ++step) {
        const int t = dir ? (SEQT - 1 - step) : step;

        // A operand (h) from LDS; pad columns 10,11 are zero -> K padding free.
        v2f a[3];
        #pragma unroll
        for (int ks = 0; ks < 3; ++ks)
            a[ks] = *(const v2f*)&h_lds[mrow * 12 + ks * 4 + 2 * half];

        // Accumulator init = input projection + biases (Din==1).
        float xv[8];
        #pragma unroll
        for (int j = 0; j < 8; ++j)
            xv[j] = x[(size_t)(b0 + j + 8 * half) * SEQT + t];

        v8f acc[3];
        #pragma unroll
        for (int tile = 0; tile < 3; ++tile)
            #pragma unroll
            for (int j = 0; j < 8; ++j)
                acc[tile][j] = xv[j] * wcol[tile] + bcol[tile];

        // Recurrence: gates += h @ whh^T  (9 fp32 WMMAs)
        #pragma unroll
        for (int tile = 0; tile < 3; ++tile)
            #pragma unroll
            for (int ks = 0; ks < 3; ++ks)
                acc[tile] = __builtin_amdgcn_wmma_f32_16x16x4_f32(
                    false, a[ks], false, bwh[tile][ks], (short)0, acc[tile],
                    false, false);

        // Dump gates to LDS in raw accumulator layout.
        #pragma unroll
        for (int tile = 0; tile < 3; ++tile)
            *(v8f*)&g_lds[lane * 24 + tile * 8] = acc[tile];

        __syncthreads();

        #pragma unroll
        for (int p = 0; p < 5; ++p) {
            const int mm = pm[p], hh = ph[p];
            const int ls = ((mm >> 3) << 4);
            const int mo = (mm & 7);
            float gi = g_lds[(((hh     ) & 15) + ls) * 24 + ((hh     ) >> 4) * 8 + mo];
            float gf = g_lds[(((hh + 10) & 15) + ls) * 24 + ((hh + 10) >> 4) * 8 + mo];
            float gg = g_lds[(((hh + 20) & 15) + ls) * 24 + ((hh + 20) >> 4) * 8 + mo];
            float go = g_lds[(((hh + 30) & 15) + ls) * 24 + ((hh + 30) >> 4) * 8 + mo];
            gi = fast_sigmoid(gi);
            gf = fast_sigmoid(gf);
            gg = fast_tanh(gg);
            go = fast_sigmoid(go);
            float c = gf * cst[p] + gi * gg;
            cst[p]  = c;
            hnew[p] = go * fast_tanh(c);
        }

        #pragma unroll
        for (int p = 0; p < 5; ++p) {
            h_lds[pm[p] * 12 + ph[p]] = hnew[p];
            outp[((size_t)(b0 + pm[p]) * SEQT + t) * HDIM + ph[p]] = hnew[p];
        }
        __syncthreads();
    }
}

// Layer-1 forward scan; input projection [16,20]@[20,40] fused as 15 extra WMMAs.
__global__ __launch_bounds__(32) void lstm_l1_fwd_scan(
    const float* __restrict__ l0f, const float* __restrict__ l0b,
    const float* __restrict__ wih,  // [40,20]
    const float* __restrict__ whh,  // [40,10]
    const float* __restrict__ bih, const float* __restrict__ bhh,
    float* __restrict__ hfinal)     // [B,10]
{
    const int b0   = blockIdx.x * 16;
    const int lane = threadIdx.x;
    const int mrow = lane & 15;
    const int half = lane >> 4;

    __shared__ __align__(16) float h_lds[16 * 12];
    __shared__ __align__(32) float g_lds[32 * 24];

    for (int i = lane; i < 16 * 12; i += 32) h_lds[i] = 0.f;

    v2f bwh[3][3], bwx[3][5];
    float bcol[3];
    #pragma unroll
    for (int tile = 0; tile < 3; ++tile) {
        int col = tile * 16 + mrow;
        int cv  = (col < GDIM);
        bcol[tile] = cv ? (bih[col] + bhh[col]) : 0.f;
        #pragma unroll
        for (int ks = 0; ks < 3; ++ks) {
            int kb = ks * 4 + 2 * half;
            v2f bb; bb.x = 0.f; bb.y = 0.f;
            if (cv && kb < HDIM)     bb.x = whh[col * HDIM + kb];
            if (cv && kb + 1 < HDIM) bb.y = whh[col * HDIM + kb + 1];
            bwh[tile][ks] = bb;
        }
        #pragma unroll
        for (int ks = 0; ks < 5; ++ks) {
            int kb = ks * 4 + 2 * half;   // 0..18, always < 20
            v2f bb; bb.x = 0.f; bb.y = 0.f;
            if (cv) { bb.x = wih[col * 20 + kb]; bb.y = wih[col * 20 + kb + 1]; }
            bwx[tile][ks] = bb;
        }
    }

    int pm[5], ph[5];
    float cst[5], hlast[5];
    #pragma unroll
    for (int p = 0; p < 5; ++p) {
        int q = lane + 32 * p;
        pm[p] = q / HDIM; ph[p] = q % HDIM;
        cst[p] = 0.f; hlast[p] = 0.f;
    }

    __syncthreads();

    for (int t = 0; t < SEQT; ++t) {
        v2f a[3];
        #pragma unroll
        for (int ks = 0; ks < 3; ++ks)
            a[ks] = *(const v2f*)&h_lds[mrow * 12 + ks * 4 + 2 * half];

        // A operand of input projection: hcat = [l0f | l0b] row (b0+mrow, t).
        v2f ax[5];
        const size_t rowbase = ((size_t)(b0 + mrow) * SEQT + t) * HDIM;
        #pragma unroll
        for (int ks = 0; ks < 5; ++ks) {
            int kb = ks * 4 + 2 * half;   // even; never straddles the 10 boundary
            const float* src = (kb < HDIM) ? (l0f + rowbase + kb)
                                           : (l0b + rowbase + (kb - HDIM));
            ax[ks] = *(const v2f*)src;
        }

        v8f acc[3];
        #pragma unroll
        for (int tile = 0; tile < 3; ++tile)
            #pragma unroll
            for (int j = 0; j < 8; ++j)
                acc[tile][j] = bcol[tile];

        #pragma unroll
        for (int tile = 0; tile < 3; ++tile) {
            #pragma unroll
            for (int ks = 0; ks < 5; ++ks)
                acc[tile] = __builtin_amdgcn_wmma_f32_16x16x4_f32(
                    false, ax[ks], false, bwx[tile][ks], (short)0, acc[tile],
                    false, false);
            #pragma unroll
            for (int ks = 0; ks < 3; ++ks)
                acc[tile] = __builtin_amdgcn_wmma_f32_16x16x4_f32(
                    false, a[ks], false, bwh[tile][ks], (short)0, acc[tile],
                    false, false);
        }

        #pragma unroll
        for (int tile = 0; tile < 3; ++tile)
            *(v8f*)&g_lds[lane * 24 + tile * 8] = acc[tile];

        __syncthreads();

        #pragma unroll
        for (int p = 0; p < 5; ++p) {
            const int mm = pm[p], hh = ph[p];
            const int ls = ((mm >> 3) << 4);
            const int mo = (mm & 7);
            float gi = g_lds[(((hh     ) & 15) + ls) * 24 + ((hh     ) >> 4) * 8 + mo];
            float gf = g_lds[(((hh + 10) & 15) + ls) * 24 + ((hh + 10) >> 4) * 8 + mo];
            float gg = g_lds[(((hh + 20) & 15) + ls) * 24 + ((hh + 20) >> 4) * 8 + mo];
            float go = g_lds[(((hh + 30) & 15) + ls) * 24 + ((hh + 30) >> 4) * 8 + mo];
            gi = fast_sigmoid(gi);
            gf = fast_sigmoid(gf);
            gg = fast_tanh(gg);
            go = fast_sigmoid(go);
            float c = gf * cst[p] + gi * gg;
            cst[p]   = c;
            hlast[p] = go * fast_tanh(c);
        }

        #pragma unroll
        for (int p = 0; p < 5; ++p)
            h_lds[pm[p] * 12 + ph[p]] = hlast[p];
        __syncthreads();
    }

    #pragma unroll
    for (int p = 0; p < 5; ++p)
        hfinal[(size_t)(b0 + pm[p]) * HDIM + ph[p]] = hlast[p];
}

// Layer-1 backward contributes only its first step at t=T-1 (h_prev=c_prev=0),
// so gates == input projection. Fuse that + ReLU + FC here, one thread per batch.
__global__ void lstm_final(
    const float* __restrict__ l0f, const float* __restrict__ l0b,
    const float* __restrict__ wih_r,                    // [40,20]
    const float* __restrict__ bih_r, const float* __restrict__ bhh_r,
    const float* __restrict__ hfinal,                   // [B,10] layer1-fwd final
    const float* __restrict__ fc_w, const float* __restrict__ fc_b,
    float* __restrict__ out)
{
    int b = blockIdx.x * blockDim.x + threadIdx.x;
    if (b >= BATCH) return;

    float hc[20];
    const size_t base = ((size_t)b * SEQT + (SEQT - 1)) * HDIM;
    #pragma unroll
    for (int k = 0; k < HDIM; ++k) {
        hc[k]        = l0f[base + k];
        hc[HDIM + k] = l0b[base + k];
    }

    float accum = fc_b[0];
    #pragma unroll
    for (int k = 0; k < HDIM; ++k)
        accum += fmaxf(hfinal[(size_t)b * HDIM + k], 0.f) * fc_w[k];

    #pragma unroll
    for (int hh = 0; hh < HDIM; ++hh) {
        float gi = bih_r[hh]      + bhh_r[hh];
        float gg = bih_r[20 + hh] + bhh_r[20 + hh];
        float go = bih_r[30 + hh] + bhh_r[30 + hh];
        #pragma unroll
        for (int k = 0; k < 20; ++k) {
            gi += wih_r[(hh     ) * 20 + k] * hc[k];
            gg += wih_r[(hh + 20) * 20 + k] * hc[k];
            go += wih_r[(hh + 30) * 20 + k] * hc[k];
        }
        float c = fast_sigmoid(gi) * fast_tanh(gg);   // f-gate * c_prev == 0
        float h = fast_sigmoid(go) * fast_tanh(c);
        accum += fmaxf(h, 0.f) * fc_w[HDIM + hh];
    }
    out[b] = accum;
}

extern "C" void kernel_launch(void* const* d_in, const int* in_sizes, int n_in,
                              void* d_out, int out_size, void* d_ws, size_t ws_size,
                              hipStream_t stream)
{
    const float* x     = (const float*)d_in[0];
    const float* wih0f = (const float*)d_in[1];
    const float* whh0f = (const float*)d_in[2];
    const float* bih0f = (const float*)d_in[3];
    const float* bhh0f = (const float*)d_in[4];
    const float* wih0r = (const float*)d_in[5];
    const float* whh0r = (const float*)d_in[6];
    const float* bih0r = (const float*)d_in[7];
    const float* bhh0r = (const float*)d_in[8];
    const float* wih1f = (const float*)d_in[9];
    const float* whh1f = (const float*)d_in[10];
    const float* bih1f = (const float*)d_in[11];
    const float* bhh1f = (const float*)d_in[12];
    const float* wih1r = (const float*)d_in[13];
    const float* bih1r = (const float*)d_in[15];
    const float* bhh1r = (const float*)d_in[16];
    const float* fc_w  = (const float*)d_in[17];
    const float* fc_b  = (const float*)d_in[18];

    float* ws   = (float*)d_ws;
    float* l0f  = ws;                                   // [B,T,10]  20 MB
    float* l0b  = l0f + (size_t)BATCH * SEQT * HDIM;    // [B,T,10]  20 MB
    float* hfin = l0b + (size_t)BATCH * SEQT * HDIM;    // [B,10]

    dim3 g0(BATCH / 16, 2);
    lstm_l0_scan<<<g0, 32, 0, stream>>>(x, wih0f, whh0f, bih0f, bhh0f,
                                        wih0r, whh0r, bih0r, bhh0r, l0f, l0b);
    lstm_l1_fwd_scan<<<BATCH / 16, 32, 0, stream>>>(l0f, l0b, wih1f, whh1f,
                                                    bih1f, bhh1f, hfin);
    lstm_final<<<(BATCH + 255) / 256, 256, 0, stream>>>(l0f, l0b, wih1r, bih1r,
                                                        bhh1r, hfin, fc_w, fc_b,
                                                        (float*)d_out);
}